// SelfAttention_45844480918334
// MI455X (gfx1250) — compile-verified
//
#include <hip/hip_runtime.h>
#include <cstdint>

// Problem constants: B=2, L=2048, DIM=2048, H=16, D=128
// M1 = B*L = 4096, N1 = 3*DIM = 6144, K = 2048

typedef float v2f __attribute__((ext_vector_type(2)));
typedef float v8f __attribute__((ext_vector_type(8)));

__device__ __forceinline__ v8f wmma4(v2f a, v2f b, v8f c) {
  // V_WMMA_F32_16X16X4_F32: D = A(16x4,f32) * B(4x16,f32) + C(16x16,f32)
  return __builtin_amdgcn_wmma_f32_16x16x4_f32(
      /*neg_a=*/false, a, /*neg_b=*/false, b,
      /*c_mod=*/(short)0, c, /*reuse_a=*/false, /*reuse_b=*/false);
}

// ---------------------------------------------------------------------------
// Kernel 1: qkv = x @ qkv_w^T   (M=4096, N=6144, K=2048)
// scatter epilogue into [3][B*H][L][D] fp32 workspace.
// ---------------------------------------------------------------------------
__global__ __launch_bounds__(128) void k_gemm_qkv(
    const float* __restrict__ x, const float* __restrict__ w,
    float* __restrict__ qkv) {
  __shared__ __align__(16) float As[64 * 16];
  __shared__ __align__(16) float Bs[64 * 16];
  const int m0 = blockIdx.y * 64;
  const int n0 = blockIdx.x * 64;
  const int tid = threadIdx.x;
  const int wave = tid >> 5, lane = tid & 31;
  const int hl = lane & 15;
  const int koff = (lane < 16) ? 0 : 2;
  const int wm = (wave >> 1) * 32;
  const int wn = (wave & 1) * 32;

  v8f acc[4] = {};

  for (int k0 = 0; k0 < 2048; k0 += 16) {
    __syncthreads();
#pragma unroll
    for (int i = 0; i < 2; ++i) {
      int idx = tid + i * 128;                 // 0..255 float4 slots
      int row = idx >> 2, c = (idx & 3) << 2;  // 64 rows x 16 cols
      *(float4*)&As[row * 16 + c] =
          *(const float4*)(x + (size_t)(m0 + row) * 2048 + k0 + c);
      *(float4*)&Bs[row * 16 + c] =
          *(const float4*)(w + (size_t)(n0 + row) * 2048 + k0 + c);
    }
    __syncthreads();
#pragma unroll
    for (int kk = 0; kk < 16; kk += 4) {
      v2f a0 = *(const v2f*)&As[(wm + hl) * 16 + kk + koff];
      v2f a1 = *(const v2f*)&As[(wm + 16 + hl) * 16 + kk + koff];
      v2f b0 = *(const v2f*)&Bs[(wn + hl) * 16 + kk + koff];
      v2f b1 = *(const v2f*)&Bs[(wn + 16 + hl) * 16 + kk + koff];
      acc[0] = wmma4(a0, b0, acc[0]);
      acc[1] = wmma4(a0, b1, acc[1]);
      acc[2] = wmma4(a1, b0, acc[2]);
      acc[3] = wmma4(a1, b1, acc[3]);
    }
  }
#pragma unroll
  for (int t = 0; t < 4; ++t) {
    const int mt = t >> 1, nt = t & 1;
#pragma unroll
    for (int r = 0; r < 8; ++r) {
      int m = m0 + wm + mt * 16 + ((lane < 16) ? r : r + 8);
      int n = n0 + wn + nt * 16 + hl;
      int which = n >> 11;        // 0:q 1:k 2:v
      int h = (n >> 7) & 15;
      int d = n & 127;
      int b = m >> 11;
      int l = m & 2047;
      qkv[(size_t)which * (size_t)8388608 +
          ((size_t)((b * 16 + h) * 2048 + l)) * 128 + d] = acc[t][r];
    }
  }
}

// ---------------------------------------------------------------------------
// Kernel 2: fused RMSNorm (over D=128) + RoPE, in place on q and k.
// One wave per (b,h,l) row; lane holds 4 consecutive d (2 RoPE pairs).
// ---------------------------------------------------------------------------
__global__ __launch_bounds__(256) void k_rmsrope(
    float* __restrict__ q, float* __restrict__ kk,
    const float* __restrict__ pe, const float* __restrict__ qs,
    const float* __restrict__ ks) {
  const int lane = threadIdx.x & 31;
  const int wid = blockIdx.x * 8 + (threadIdx.x >> 5);
  const int nrows = 65536;  // B*H*L
  const int isk = wid >= nrows;
  const int row = isk ? wid - nrows : wid;
  float* base = (isk ? kk : q) + (size_t)row * 128;
  const float* sc = isk ? ks : qs;
  const int l = row & 2047;

  float4 xv = *(const float4*)(base + lane * 4);
  float ss = xv.x * xv.x + xv.y * xv.y + xv.z * xv.z + xv.w * xv.w;
#pragma unroll
  for (int sh = 16; sh >= 1; sh >>= 1) ss += __shfl_xor(ss, sh, 32);
  float rrms = rsqrtf(ss * (1.0f / 128.0f) + 1e-6f);
  float4 s4 = *(const float4*)(sc + lane * 4);
  float x0 = xv.x * rrms * s4.x;
  float x1 = xv.y * rrms * s4.y;
  float x2 = xv.z * rrms * s4.z;
  float x3 = xv.w * rrms * s4.w;
  // pe: [L][64][2][2] -> 256 floats per token; lane owns pairs 2*lane, 2*lane+1
  const float* peb = pe + (size_t)l * 256 + lane * 8;
  float4 p0 = *(const float4*)(peb);       // pair 2*lane:   (00,01,10,11)
  float4 p1 = *(const float4*)(peb + 4);   // pair 2*lane+1
  float4 yv;
  yv.x = p0.x * x0 + p0.y * x1;
  yv.y = p0.z * x0 + p0.w * x1;
  yv.z = p1.x * x2 + p1.y * x3;
  yv.w = p1.z * x2 + p1.w * x3;
  *(float4*)(base + lane * 4) = yv;
}

// ---------------------------------------------------------------------------
// Kernel 3: flash attention per head. grid=(L/64, B*H), 4 waves/block.
// Each wave: 16 query rows, full D=128 output, online softmax over 64-key
// blocks. Q in registers; K row-major + V transposed in LDS; P transposed
// through per-wave LDS region (overlaid on K buffer, barrier-guarded).
// ---------------------------------------------------------------------------
__global__ __launch_bounds__(128) void k_attn(
    const float* __restrict__ q, const float* __restrict__ k,
    const float* __restrict__ v, float* __restrict__ o) {
  __shared__ __align__(16) float Ks[64 * 128];  // [key][d]; reused as P
  __shared__ __align__(16) float Vt[128 * 64];  // [d][key]
  const int bh = blockIdx.y;
  const int q0 = blockIdx.x * 64;
  const int tid = threadIdx.x;
  const int wave = tid >> 5, lane = tid & 31;
  const int hl = lane & 15;
  const int koff = (lane < 16) ? 0 : 2;
  const float rscale = 0.08838834764831845f;  // 1/sqrt(128)

  const float* qb = q + (size_t)bh * (2048 * 128);
  const float* kbase = k + (size_t)bh * (2048 * 128);
  const float* vbase = v + (size_t)bh * (2048 * 128);

  // Q fragments: 32 K-slices of the wave's 16 rows
  v2f qa[32];
  {
    const float* qrow = qb + (size_t)(q0 + wave * 16 + hl) * 128;
#pragma unroll
    for (int s = 0; s < 32; ++s) qa[s] = *(const v2f*)(qrow + 4 * s + koff);
  }

  v8f oacc[8] = {};
  float m_i[8], l_i[8];
#pragma unroll
  for (int r = 0; r < 8; ++r) { m_i[r] = -3.4e38f; l_i[r] = 0.f; }

  for (int kb0 = 0; kb0 < 2048; kb0 += 64) {
    __syncthreads();  // prior P / Vt consumers done -> safe to restage
    for (int i = tid; i < 64 * 32; i += 128) {
      int row = i >> 5;
      int c4 = (i & 31) << 2;
      *(float4*)&Ks[row * 128 + c4] =
          *(const float4*)(kbase + (size_t)(kb0 + row) * 128 + c4);
      float4 vv = *(const float4*)(vbase + (size_t)(kb0 + row) * 128 + c4);
      Vt[(c4 + 0) * 64 + row] = vv.x;
      Vt[(c4 + 1) * 64 + row] = vv.y;
      Vt[(c4 + 2) * 64 + row] = vv.z;
      Vt[(c4 + 3) * 64 + row] = vv.w;
    }
    __syncthreads();

    // S = Q * K^T : 16x64 in four 16x16 tiles
    v8f sacc[4] = {};
#pragma unroll
    for (int nt = 0; nt < 4; ++nt) {
      const float* krow = &Ks[(nt * 16 + hl) * 128];
      v8f a = sacc[nt];
#pragma unroll
      for (int s = 0; s < 32; ++s)
        a = wmma4(qa[s], *(const v2f*)(krow + 4 * s + koff), a);
      sacc[nt] = a;
    }

    // online softmax, row r lives in this lane's 16-lane group
#pragma unroll
    for (int r = 0; r < 8; ++r) {
      float mx = -3.4e38f;
#pragma unroll
      for (int nt = 0; nt < 4; ++nt) {
        sacc[nt][r] *= rscale;
        mx = fmaxf(mx, sacc[nt][r]);
      }
#pragma unroll
      for (int sh = 8; sh >= 1; sh >>= 1) mx = fmaxf(mx, __shfl_xor(mx, sh, 32));
      float mnew = fmaxf(m_i[r], mx);
      float alpha = __expf(m_i[r] - mnew);
      m_i[r] = mnew;
      float rs = 0.f;
#pragma unroll
      for (int nt = 0; nt < 4; ++nt) {
        float p = __expf(sacc[nt][r] - mnew);
        sacc[nt][r] = p;
        rs += p;
      }
#pragma unroll
      for (int sh = 8; sh >= 1; sh >>= 1) rs += __shfl_xor(rs, sh, 32);
      l_i[r] = l_i[r] * alpha + rs;
#pragma unroll
      for (int nd = 0; nd < 8; ++nd) oacc[nd][r] *= alpha;
    }

    __syncthreads();  // all waves finished reading Ks -> overlay P on it
    float* Pw = &Ks[wave * (16 * 64)];
#pragma unroll
    for (int nt = 0; nt < 4; ++nt) {
#pragma unroll
      for (int r = 0; r < 8; ++r) {
        int row = (lane < 16) ? r : r + 8;
        Pw[row * 64 + nt * 16 + hl] = sacc[nt][r];
      }
    }
    // reload P as A fragments (wave-private region; DScnt ordering suffices)
    v2f pa[16];
    {
      const float* prow = &Pw[hl * 64];
#pragma unroll
      for (int s = 0; s < 16; ++s) pa[s] = *(const v2f*)(prow + 4 * s + koff);
    }
    // O += P * V
#pragma unroll
    for (int nd = 0; nd < 8; ++nd) {
      const float* vtrow = &Vt[(nd * 16 + hl) * 64];
      v8f a = oacc[nd];
#pragma unroll
      for (int s = 0; s < 16; ++s)
        a = wmma4(pa[s], *(const v2f*)(vtrow + 4 * s + koff), a);
      oacc[nd] = a;
    }
  }

  // normalize + store to o[b][t][h*128+d] (token-major for final GEMM)
  const int b = bh >> 4, h = bh & 15;
#pragma unroll
  for (int nd = 0; nd < 8; ++nd) {
#pragma unroll
    for (int r = 0; r < 8; ++r) {
      int row = (lane < 16) ? r : r + 8;
      int t = q0 + wave * 16 + row;
      int d = nd * 16 + hl;
      o[((size_t)(b * 2048 + t)) * 2048 + h * 128 + d] = oacc[nd][r] / l_i[r];
    }
  }
}

// ---------------------------------------------------------------------------
// Kernel 4: out = o @ proj_w^T + bias   (M=4096, N=2048, K=2048)
// ---------------------------------------------------------------------------
__global__ __launch_bounds__(128) void k_gemm_out(
    const float* __restrict__ ov, const float* __restrict__ w,
    const float* __restrict__ bias, float* __restrict__ out) {
  __shared__ __align__(16) float As[64 * 16];
  __shared__ __align__(16) float Bs[64 * 16];
  const int m0 = blockIdx.y * 64;
  const int n0 = blockIdx.x * 64;
  const int tid = threadIdx.x;
  const int wave = tid >> 5, lane = tid & 31;
  const int hl = lane & 15;
  const int koff = (lane < 16) ? 0 : 2;
  const int wm = (wave >> 1) * 32;
  const int wn = (wave & 1) * 32;

  v8f acc[4] = {};

  for (int k0 = 0; k0 < 2048; k0 += 16) {
    __syncthreads();
#pragma unroll
    for (int i = 0; i < 2; ++i) {
      int idx = tid + i * 128;
      int row = idx >> 2, c = (idx & 3) << 2;
      *(float4*)&As[row * 16 + c] =
          *(const float4*)(ov + (size_t)(m0 + row) * 2048 + k0 + c);
      *(float4*)&Bs[row * 16 + c] =
          *(const float4*)(w + (size_t)(n0 + row) * 2048 + k0 + c);
    }
    __syncthreads();
#pragma unroll
    for (int kk = 0; kk < 16; kk += 4) {
      v2f a0 = *(const v2f*)&As[(wm + hl) * 16 + kk + koff];
      v2f a1 = *(const v2f*)&As[(wm + 16 + hl) * 16 + kk + koff];
      v2f b0 = *(const v2f*)&Bs[(wn + hl) * 16 + kk + koff];
      v2f b1 = *(const v2f*)&Bs[(wn + 16 + hl) * 16 + kk + koff];
      acc[0] = wmma4(a0, b0, acc[0]);
      acc[1] = wmma4(a0, b1, acc[1]);
      acc[2] = wmma4(a1, b0, acc[2]);
      acc[3] = wmma4(a1, b1, acc[3]);
    }
  }
#pragma unroll
  for (int t = 0; t < 4; ++t) {
    const int mt = t >> 1, nt = t & 1;
#pragma unroll
    for (int r = 0; r < 8; ++r) {
      int m = m0 + wm + mt * 16 + ((lane < 16) ? r : r + 8);
      int n = n0 + wn + nt * 16 + hl;
      out[(size_t)m * 2048 + n] = acc[t][r] + bias[n];
    }
  }
}

// ---------------------------------------------------------------------------
extern "C" void kernel_launch(void* const* d_in, const int* in_sizes, int n_in,
                              void* d_out, int out_size, void* d_ws,
                              size_t ws_size, hipStream_t stream) {
  const float* x = (const float*)d_in[0];
  const float* pe = (const float*)d_in[1];
  const float* qkv_w = (const float*)d_in[2];
  const float* q_scale = (const float*)d_in[3];
  const float* k_scale = (const float*)d_in[4];
  const float* proj_w = (const float*)d_in[5];
  const float* proj_b = (const float*)d_in[6];
  float* out = (float*)d_out;

  float* ws = (float*)d_ws;
  const size_t per = (size_t)8388608;  // B*H*L*D
  float* qw = ws;                      // q [B*H][L][D]
  float* kw = ws + per;                // k
  float* vw = ws + 2 * per;            // v
  float* ow = ws + 3 * per;            // attn out [B][L][DIM]

  // qkv projection (writes q/k/v scattered into head-major layout)
  k_gemm_qkv<<<dim3(96, 64), 128, 0, stream>>>(x, qkv_w, qw);
  // fused RMSNorm + RoPE on q and k (131072 rows, 8 waves per block)
  k_rmsrope<<<dim3(16384), 256, 0, stream>>>(qw, kw, pe, q_scale, k_scale);
  // flash attention per (head, 64-query block)
  k_attn<<<dim3(32, 32), 128, 0, stream>>>(qw, kw, vw, ow);
  // output projection + bias
  k_gemm_out<<<dim3(32, 64), 128, 0, stream>>>(ow, proj_w, proj_b, out);
}